// lossFunc_81166291960473
// MI455X (gfx1250) — compile-verified
//
#include <hip/hip_runtime.h>
#include <stdint.h>

#ifndef __has_builtin
#define __has_builtin(x) 0
#endif

#if defined(__AMDGCN__) && __has_builtin(__builtin_amdgcn_global_load_async_to_lds_b128)
#define HAVE_ASYNC_LDS 1
#else
#define HAVE_ASYNC_LDS 0
#endif

// Problem constants (B,T,Q) = (64, 200, 4096)
constexpr int kB    = 64;
constexpr int kT    = 200;
constexpr int kQ    = 4096;
constexpr int kTm1  = 199;
constexpr int kRow  = 2 * kQ;        // 8192 floats per batch row
constexpr int kRows = kB * kTm1;     // 12736 (b,t) pairs

// Types matching the async-to-LDS builtin's parameters:
//   arg0: int __attribute__((vector_size(16))) addrspace(1)*  (global src)
//   arg1: same pointee in addrspace(3)                        (LDS dst)
typedef int v4i __attribute__((vector_size(16)));
typedef __attribute__((address_space(1))) v4i GV4;   // global int4*
typedef __attribute__((address_space(3))) v4i LV4;   // LDS int4*

// ---------------------------------------------------------------------------
// Kernel 1: one block per (b,t).  Stream batch[b,t+1,:8192] (async copy to
// LDS on CDNA5), reduce to {s_cor, s_inc, nonzero index j, nonzero value},
// then gather the single needed pred element:  p = val * pred[b,t, j mod Q].
// ---------------------------------------------------------------------------
__global__ __launch_bounds__(256) void rowreduce_kernel(
    const float* __restrict__ pred, const float* __restrict__ batch,
    float* __restrict__ p_raw, float* __restrict__ a_raw,
    float* __restrict__ flag_arr)
{
  const int n   = blockIdx.x;
  const int b   = n / kTm1;
  const int t   = n - b * kTm1;           // 0..198 ; uses batch row t+1
  const int tid = threadIdx.x;

  const float* grow = batch + ((size_t)b * kT + (t + 1)) * (size_t)kRow;

  __shared__ alignas(16) float rowbuf[kRow];     // 32 KB staging buffer
  __shared__ float r_scor[256], r_sinc[256], r_val[256];
  __shared__ int   r_j[256];

#if HAVE_ASYNC_LDS
  // Async-stage the whole 32 KB row into LDS: 8 x (256 lanes x b128).
  for (int k = 0; k < 8; ++k) {
    const int c = tid + (k << 8);                         // float4 chunk id
    GV4* g = (GV4*)(unsigned long long)(grow + (size_t)c * 4);
    LV4* l = (LV4*)(unsigned int)(unsigned long long)(rowbuf + c * 4);
    __builtin_amdgcn_global_load_async_to_lds_b128(g, l, 0, 0);
  }
#if __has_builtin(__builtin_amdgcn_s_wait_asynccnt)
  __builtin_amdgcn_s_wait_asynccnt(0);
#else
  asm volatile("s_wait_asynccnt 0" ::: "memory");
#endif
  __syncthreads();
#endif

  float scor = 0.f, sinc = 0.f, val = 0.f;
  int   jidx = -1;
  for (int k = 0; k < 8; ++k) {
    const int c = tid + (k << 8);
#if HAVE_ASYNC_LDS
    const float4 v = ((const float4*)rowbuf)[c];
#else
    const float4 v = ((const float4*)grow)[c];
#endif
    const float s = (v.x + v.y) + (v.z + v.w);
    if (c < 1024) scor += s; else sinc += s;   // first 4096 floats = cor half
    if (s != 0.f) {                            // at most one nonzero per row
      const int base = c << 2;
      if      (v.x != 0.f) { jidx = base;     val = v.x; }
      else if (v.y != 0.f) { jidx = base + 1; val = v.y; }
      else if (v.z != 0.f) { jidx = base + 2; val = v.z; }
      else                 { jidx = base + 3; val = v.w; }
    }
  }

  r_scor[tid] = scor; r_sinc[tid] = sinc; r_val[tid] = val; r_j[tid] = jidx;
  __syncthreads();
  for (int s = 128; s > 0; s >>= 1) {
    if (tid < s) {
      r_scor[tid] += r_scor[tid + s];
      r_sinc[tid] += r_sinc[tid + s];
      r_val[tid]  += r_val[tid + s];
      r_j[tid]     = (r_j[tid] > r_j[tid + s]) ? r_j[tid] : r_j[tid + s];
    }
    __syncthreads();
  }

  if (tid == 0) {
    const float sc = r_scor[0], si = r_sinc[0];
    const int   j  = r_j[0];
    float pv = 0.f;
    if (j >= 0)   // p = dot(pred_row, one_hot) == single gathered element
      pv = r_val[0] * pred[((size_t)b * kT + t) * (size_t)kQ + (j & (kQ - 1))];
    p_raw[n]    = pv;
    a_raw[n]    = floorf((sc - si + 1.f) * 0.5f);   // ((cor-inc+1)//2)
    flag_arr[n] = sc + si;
  }
}

// ---------------------------------------------------------------------------
// Kernel 2: one block per batch b.  last/start/mask/cnt + clamped BCE,
// writes the three (64,199) outputs and a per-b partial loss.
// ---------------------------------------------------------------------------
__global__ __launch_bounds__(256) void finalize_kernel(
    const float* __restrict__ p_raw, const float* __restrict__ a_raw,
    const float* __restrict__ flag_arr, const int* __restrict__ isTest,
    const int* __restrict__ testseqlen, float* __restrict__ out,
    float* __restrict__ partial)
{
  const int b = blockIdx.x, tid = threadIdx.x;
  __shared__ int   r_last[256];
  __shared__ float r_sum[256], r_cnt[256];

  const int  n  = b * kTm1 + tid;
  const bool in = (tid < kTm1);

  const float fl = in ? flag_arr[n] : 0.f;
  r_last[tid] = (in && fl > 0.f) ? tid : -1;
  __syncthreads();
  for (int s = 128; s > 0; s >>= 1) {
    if (tid < s) r_last[tid] = max(r_last[tid], r_last[tid + s]);
    __syncthreads();
  }
  const int last = max(r_last[0], 0);

  int start = 0;
  if (isTest[0]) {
    const int len = last + 1, tsl = testseqlen[0];
    start = (len > tsl) ? (len - tsl) : 0;
  }

  const float maskf = (in && tid >= start && tid <= last) ? 1.f : 0.f;
  const float p = in ? p_raw[n] : 0.f;
  const float a = in ? a_raw[n] : 0.f;

  const float logp = fmaxf(logf(p),     -100.f);   // log(0) = -inf -> -100
  const float lo1  = fmaxf(log1pf(-p),  -100.f);
  const float bce  = -(a * logp + (1.f - a) * lo1);

  if (in) {
    out[1 + n]             = p * maskf;   // p * maskf   (64,199)
    out[1 + kRows + n]     = a * maskf;   // a * maskf   (64,199)
    out[1 + 2 * kRows + n] = maskf;       // mask        (64,199)
  }

  r_sum[tid] = bce * maskf;
  r_cnt[tid] = maskf;
  __syncthreads();
  for (int s = 128; s > 0; s >>= 1) {
    if (tid < s) { r_sum[tid] += r_sum[tid + s]; r_cnt[tid] += r_cnt[tid + s]; }
    __syncthreads();
  }
  if (tid == 0) partial[b] = r_sum[0] / r_cnt[0];
}

// ---------------------------------------------------------------------------
// Kernel 3: sum the 64 per-batch partial losses -> out[0].
// ---------------------------------------------------------------------------
__global__ __launch_bounds__(64) void sumloss_kernel(
    const float* __restrict__ partial, float* __restrict__ out)
{
  const int tid = threadIdx.x;
  __shared__ float s[64];
  s[tid] = partial[tid];
  __syncthreads();
  for (int k = 32; k > 0; k >>= 1) {
    if (tid < k) s[tid] += s[tid + k];
    __syncthreads();
  }
  if (tid == 0) out[0] = s[0];
}

// ---------------------------------------------------------------------------
extern "C" void kernel_launch(void* const* d_in, const int* in_sizes, int n_in,
                              void* d_out, int out_size, void* d_ws, size_t ws_size,
                              hipStream_t stream)
{
  const float* pred       = (const float*)d_in[0];   // (64,200,4096) f32
  const float* batch      = (const float*)d_in[1];   // (64,200,8192) f32
  const int*   isTest     = (const int*)d_in[2];     // scalar
  const int*   testseqlen = (const int*)d_in[3];     // scalar
  float*       out        = (float*)d_out;           // 1 + 3*12736 floats

  float* p_raw   = (float*)d_ws;          // 12736
  float* a_raw   = p_raw + kRows;         // 12736
  float* flag    = a_raw + kRows;         // 12736
  float* partial = flag  + kRows;         // 64

  rowreduce_kernel<<<kRows, 256, 0, stream>>>(pred, batch, p_raw, a_raw, flag);
  finalize_kernel<<<kB, 256, 0, stream>>>(p_raw, a_raw, flag, isTest,
                                          testseqlen, out, partial);
  sumloss_kernel<<<1, 64, 0, stream>>>(partial, out);
}